// GATLayer_53249004536214
// MI455X (gfx1250) — compile-verified
//
#include <hip/hip_runtime.h>
#include <math.h>

typedef float v2f __attribute__((ext_vector_type(2)));
typedef float v8f __attribute__((ext_vector_type(8)));

constexpr int kNodes = 50000;
constexpr int kEdges = 800000;
constexpr int kDim   = 128;

// Workspace layout (floats): h | s_src | s_dst | e | gmax(u32) | gsum
constexpr size_t H_OFF    = 0;
constexpr size_t SSRC_OFF = (size_t)kNodes * kDim;   // 6,400,000
constexpr size_t SDST_OFF = SSRC_OFF + kNodes;       // 6,450,000
constexpr size_t E_OFF    = SDST_OFF + kNodes;       // 6,500,000
constexpr size_t GMAX_OFF = E_OFF + kEdges;          // 7,300,000
constexpr size_t GSUM_OFF = GMAX_OFF + 1;            // total ~29.2 MB

__device__ __forceinline__ unsigned f2ord(float f) {
  unsigned u = __float_as_uint(f);
  return (u & 0x80000000u) ? ~u : (u | 0x80000000u);
}
__device__ __forceinline__ float ord2f(unsigned u) {
  unsigned v = (u & 0x80000000u) ? (u & 0x7FFFFFFFu) : ~u;
  return __uint_as_float(v);
}

// ---------------- TDM (Tensor Data Mover) async global->LDS -----------------
#if __has_builtin(__builtin_amdgcn_tensor_load_to_lds) && \
    __has_builtin(__builtin_amdgcn_s_wait_tensorcnt)
#define USE_TDM 1

typedef unsigned tdm_v4u __attribute__((ext_vector_type(4)));
typedef int      tdm_v4i __attribute__((ext_vector_type(4)));
typedef int      tdm_v8i __attribute__((ext_vector_type(8)));

// 2D tile load: dim0 = row length (elements), dim1 = #rows available (OOB rows
// zero-fill), tile0 x tile1 = tile shape, stride0 = elements between rows.
// D# bitfields per CDNA5 ISA ch.8 (data_size=2 -> 4-byte elements, type=2).
__device__ __forceinline__ void tdm_load_2d(unsigned lds_off, const void* gsrc,
                                            unsigned dim0, unsigned dim1,
                                            unsigned tile0, unsigned tile1,
                                            unsigned stride0) {
  const unsigned long long ga = (unsigned long long)(uintptr_t)gsrc;
  tdm_v4u g0;
  g0.x = 0x1u;                                        // count=1, user mode
  g0.y = lds_off;                                     // lds_addr [63:32]
  g0.z = (unsigned)(ga & 0xFFFFFFFFu);                // global_addr[31:0]
  g0.w = (unsigned)((ga >> 32) & 0x01FFFFFFu)         // global_addr[56:32]
         | 0x80000000u;                               // type=2 ("image")
  tdm_v8i g1;
  g1[0] = (int)(2u << 16);                            // data_size=4B; mask=0
  g1[1] = (int)(dim0 << 16);                          // tensor_dim0[15:0]
  g1[2] = (int)((dim0 >> 16) | (dim1 << 16));         // dim0 hi | dim1 lo
  g1[3] = (int)((dim1 >> 16) | (tile0 << 16));        // dim1 hi | tile_dim0
  g1[4] = (int)(tile1 & 0xFFFFu);                     // tile_dim1; tile_dim2=0
  g1[5] = (int)stride0;                               // tensor_dim0_stride lo
  g1[6] = 0;                                          // stride0 hi | stride1 lo
  g1[7] = 0;
  const tdm_v4i z4 = {0, 0, 0, 0};
#if __clang_major__ >= 23
  const tdm_v8i z8 = {0, 0, 0, 0, 0, 0, 0, 0};
  __builtin_amdgcn_tensor_load_to_lds(g0, g1, z4, z4, z8, 0);
#else
  __builtin_amdgcn_tensor_load_to_lds(g0, g1, z4, z4, 0);
#endif
}
#endif  // USE_TDM

// ---------------- init: zero output, seed softmax scalars -------------------
__global__ void k_init(float* __restrict__ out, float* __restrict__ ws) {
  size_t i = (size_t)blockIdx.x * blockDim.x + threadIdx.x;
  out[i] = 0.0f;                       // grid sized exactly to kNodes*kDim
  if (i == 0) {
    *(unsigned*)(ws + GMAX_OFF) = f2ord(-INFINITY);
    ws[GSUM_OFF] = 0.0f;
  }
}

// ---------------- h = x @ W + b via v_wmma_f32_16x16x4_f32 ------------------
// 256 threads = 8 wave32s. Block covers a 64-row slab; wave w owns N-tile
// [16w, 16w+16). B(W) fragments are hoisted to registers (loop-invariant
// across the 4 M-subtiles); K=128 in steps of 4.
__global__ void k_gemm(const float* __restrict__ x, const float* __restrict__ Ww,
                       const float* __restrict__ Wb, float* __restrict__ h) {
  extern __shared__ float smem[];
  float* sW = smem;                    // 128*128 floats
  float* sX = smem + kDim * kDim;      // 64*128 floats
  float* sB = sX + 64 * kDim;          // 128 floats

  const int t  = threadIdx.x;
  const int m0 = blockIdx.x * 64;

#ifdef USE_TDM
  const int rows = (kNodes - m0 < 64) ? (kNodes - m0) : 64;
  if (t < 32) {                        // wave 0 issues both DMA descriptors
    tdm_load_2d((unsigned)(uintptr_t)sW, Ww, kDim, kDim, kDim, kDim, kDim);
    tdm_load_2d((unsigned)(uintptr_t)sX, x + (size_t)m0 * kDim,
                kDim, (unsigned)rows, kDim, 64, kDim);   // OOB rows -> 0
    __builtin_amdgcn_s_wait_tensorcnt(0);
  }
  if (t < kDim) sB[t] = Wb[t];
  __syncthreads();
#else
  for (int i = t; i < (kDim * kDim) / 4; i += 256)
    ((float4*)sW)[i] = ((const float4*)Ww)[i];
  for (int i = t; i < (64 * kDim) / 4; i += 256) {
    int r = i >> 5, c4 = i & 31;
    int gr = m0 + r; if (gr >= kNodes) gr = kNodes - 1;   // clamp tail reads
    ((float4*)(sX + r * kDim))[c4] = ((const float4*)(x + (size_t)gr * kDim))[c4];
  }
  if (t < kDim) sB[t] = Wb[t];
  __syncthreads();
#endif

  const int lane = t & 31;
  const int wave = t >> 5;
  const int n0   = wave * 16;
  const int l16  = lane & 15;
  const int kh   = lane >> 4;          // lanes 16-31 hold K+2,K+3

  // Hoist all B fragments for this wave's N-tile: 32 k-steps x v2f = 64 VGPRs.
  v2f bfrag[32];
#pragma unroll
  for (int kk = 0; kk < 32; ++kk) {
    const int ka = kk * 4 + kh * 2;
    bfrag[kk].x = sW[(size_t)ka * kDim + n0 + l16];
    bfrag[kk].y = sW[(size_t)(ka + 1) * kDim + n0 + l16];
  }

  const float bias = sB[n0 + l16];
  const int gcol = n0 + l16;

  for (int mt = 0; mt < 4; ++mt) {
    v8f acc = {0.f, 0.f, 0.f, 0.f, 0.f, 0.f, 0.f, 0.f};
    const float* xr = sX + (size_t)(mt * 16 + l16) * kDim + kh * 2;
#pragma unroll
    for (int kk = 0; kk < 32; ++kk) {
      const v2f a = *(const v2f*)(xr + kk * 4);   // ds_load_b64
      acc = __builtin_amdgcn_wmma_f32_16x16x4_f32(
          /*neg_a=*/false, a, /*neg_b=*/false, bfrag[kk],
          /*c_mod=*/(short)0, acc, /*reuse_a=*/false, /*reuse_b=*/false);
    }
    const int gr0 = m0 + mt * 16 + kh * 8;
#pragma unroll
    for (int r = 0; r < 8; ++r) {
      int gr = gr0 + r;
      if (gr < kNodes) h[(size_t)gr * kDim + gcol] = acc[r] + bias;
    }
  }
}

// ------------- per-node attention scalars: s=h·a_src, t=h·a_dst -------------
__global__ void k_scores(const float* __restrict__ h, const float* __restrict__ aw,
                         float* __restrict__ ssrc, float* __restrict__ sdst) {
  const int lane = threadIdx.x & 31;
  const int node = blockIdx.x * 8 + (threadIdx.x >> 5);
  const float4 hv = ((const float4*)(h + (size_t)node * kDim))[lane];
  const float4 as = ((const float4*)aw)[lane];
  const float4 ad = ((const float4*)(aw + kDim))[lane];
  float s = hv.x * as.x + hv.y * as.y + hv.z * as.z + hv.w * as.w;
  float d = hv.x * ad.x + hv.y * ad.y + hv.z * ad.z + hv.w * ad.w;
  for (int m = 16; m; m >>= 1) {
    s += __shfl_xor(s, m, 32);
    d += __shfl_xor(d, m, 32);
  }
  if (lane == 0) { ssrc[node] = s; sdst[node] = d; }
}

// ------------- edge logits + leaky_relu + global max (pass 1) ---------------
__global__ void k_logits(const int* __restrict__ row, const int* __restrict__ col,
                         const float* __restrict__ ssrc, const float* __restrict__ sdst,
                         const float* __restrict__ ab, float* __restrict__ e,
                         float* __restrict__ ws) {
  __shared__ float red[8];
  const int k = blockIdx.x * 256 + threadIdx.x;     // exact multiple, no tail
  float v = ssrc[row[k]] + sdst[col[k]] + ab[0];
  v = v > 0.0f ? v : 0.01f * v;
  e[k] = v;
  float m = v;
  for (int s = 16; s; s >>= 1) m = fmaxf(m, __shfl_xor(m, s, 32));
  if ((threadIdx.x & 31) == 0) red[threadIdx.x >> 5] = m;
  __syncthreads();
  if (threadIdx.x == 0) {
    float bm = red[0];
#pragma unroll
    for (int i = 1; i < 8; ++i) bm = fmaxf(bm, red[i]);
    atomicMax((unsigned*)(ws + GMAX_OFF), f2ord(bm));
  }
}

// ------------- exp(e - max) + global sum (pass 2) ---------------------------
__global__ void k_expsum(float* __restrict__ e, float* __restrict__ ws) {
  __shared__ float red[8];
  const int k = blockIdx.x * 256 + threadIdx.x;
  const float gmax = ord2f(*(const unsigned*)(ws + GMAX_OFF));
  float w = __expf(e[k] - gmax);
  e[k] = w;
  for (int s = 16; s; s >>= 1) w += __shfl_xor(w, s, 32);
  if ((threadIdx.x & 31) == 0) red[threadIdx.x >> 5] = w;
  __syncthreads();
  if (threadIdx.x == 0) {
    float bs = 0.0f;
#pragma unroll
    for (int i = 0; i < 8; ++i) bs += red[i];
    atomicAdd(ws + GSUM_OFF, bs);
  }
}

// ------------- out[row] += attn * h[col]; one wave per edge -----------------
__global__ void k_scatter(const int* __restrict__ row, const int* __restrict__ col,
                          const float* __restrict__ e, const float* __restrict__ h,
                          const float* __restrict__ ws, float* __restrict__ out) {
  const int lane = threadIdx.x & 31;
  const int k = blockIdx.x * 8 + (threadIdx.x >> 5);
  const float attn = e[k] * (1.0f / ws[GSUM_OFF]);
  const int r = row[k], c = col[k];
  const float4 hv = ((const float4*)(h + (size_t)c * kDim))[lane];
  float* o = out + (size_t)r * kDim + lane * 4;
  atomicAdd(o + 0, attn * hv.x);
  atomicAdd(o + 1, attn * hv.y);
  atomicAdd(o + 2, attn * hv.z);
  atomicAdd(o + 3, attn * hv.w);
}

extern "C" void kernel_launch(void* const* d_in, const int* in_sizes, int n_in,
                              void* d_out, int out_size, void* d_ws, size_t ws_size,
                              hipStream_t stream) {
  const float* x  = (const float*)d_in[0];
  const int*   ei = (const int*)d_in[1];     // [2][kEdges], int32
  const float* Ww = (const float*)d_in[2];
  const float* Wb = (const float*)d_in[3];
  const float* aw = (const float*)d_in[4];   // [256]: a_src | a_dst
  const float* ab = (const float*)d_in[5];
  float* out = (float*)d_out;
  float* ws  = (float*)d_ws;

  const int* row = ei;
  const int* col = ei + kEdges;
  float* h    = ws + H_OFF;
  float* ssrc = ws + SSRC_OFF;
  float* sdst = ws + SDST_OFF;
  float* e    = ws + E_OFF;

  k_init<<<(kNodes * kDim) / 256, 256, 0, stream>>>(out, ws);

  const size_t smem = (size_t)(kDim * kDim + 64 * kDim + kDim) * sizeof(float);
  k_gemm<<<(kNodes + 63) / 64, 256, smem, stream>>>(x, Ww, Wb, h);

  k_scores<<<kNodes / 8, 256, 0, stream>>>(h, aw, ssrc, sdst);
  k_logits<<<kEdges / 256, 256, 0, stream>>>(row, col, ssrc, sdst, ab, e, ws);
  k_expsum<<<kEdges / 256, 256, 0, stream>>>(e, ws);
  k_scatter<<<kEdges / 8, 256, 0, stream>>>(row, col, e, h, ws, out);
}